// LIGHT_69441031242585
// MI455X (gfx1250) — compile-verified
//
#include <hip/hip_runtime.h>
#include <stdint.h>

#define N_USER  100000
#define N_ITEM  50000
#define N_NODES 150000
#define NNZ     3000000
#define EMB     64
#define N_LAYERS 3
#define BATCH   4096

#define CHUNK 1024          // nnz staged per block in scatter
#define NP1   (N_NODES + 1) // scan length (150001)
#define NBLK  ((NP1 + 1023) / 1024)  // 147 scan blocks

#if defined(__gfx1250__) && __has_builtin(__builtin_amdgcn_global_load_async_to_lds_b128)
#define ASYNC_LDS 1
#else
#define ASYNC_LDS 0
#endif

#if defined(__gfx1250__) && __has_builtin(__builtin_amdgcn_wmma_f32_16x16x4_f32)
#define HAVE_WMMA 1
#else
#define HAVE_WMMA 0
#endif

typedef int v4i __attribute__((ext_vector_type(4)));
typedef __attribute__((address_space(1))) v4i GV4;  // global-side 16B granule
typedef __attribute__((address_space(3))) v4i LV4;  // LDS-side 16B granule
typedef float v2f __attribute__((ext_vector_type(2)));
typedef float v8f __attribute__((ext_vector_type(8)));

// ---------------------------------------------------------------------------
// acc = cur = concat(user_emb, item_emb); float4 per thread
// ---------------------------------------------------------------------------
__global__ void lg_init(const float* __restrict__ ue, const float* __restrict__ ie,
                        float* __restrict__ acc, float* __restrict__ cur) {
  long long g = (long long)blockIdx.x * blockDim.x + threadIdx.x;  // float4 idx
  long long e = g * 4;
  if (e >= (long long)N_NODES * EMB) return;
  int node = (int)(e >> 6);
  int w    = (int)(e & 63);
  const float* src = (node < N_USER)
                         ? (ue + (long long)node * EMB + w)
                         : (ie + (long long)(node - N_USER) * EMB + w);
  float4 v = *(const float4*)src;
  *(float4*)(acc + e) = v;
  *(float4*)(cur + e) = v;
}

// ---------------------------------------------------------------------------
// CSR build step 1: row histogram into rowptr[1 + r]
// ---------------------------------------------------------------------------
__global__ void lg_hist(const int* __restrict__ rows, int* __restrict__ rowptr) {
  int i = blockIdx.x * blockDim.x + threadIdx.x;
  if (i >= NNZ) return;
  atomicAdd(&rowptr[1 + rows[i]], 1);
}

// ---------------------------------------------------------------------------
// CSR build step 2a: per-block inclusive scan (1024 elems/block, Hillis-Steele)
// ---------------------------------------------------------------------------
__global__ void lg_scanA(int* __restrict__ rowptr, int* __restrict__ blksums) {
  __shared__ int sh[1024];
  int tid = threadIdx.x;
  int i = blockIdx.x * 1024 + tid;
  int x = (i < NP1) ? rowptr[i] : 0;
  sh[tid] = x;
  __syncthreads();
  for (int offs = 1; offs < 1024; offs <<= 1) {
    int y = (tid >= offs) ? sh[tid - offs] : 0;
    __syncthreads();
    sh[tid] += y;
    __syncthreads();
  }
  if (i < NP1) rowptr[i] = sh[tid];
  if (tid == 1023) blksums[blockIdx.x] = sh[1023];
}

// ---------------------------------------------------------------------------
// CSR build step 2b: exclusive scan of the 147 block sums (tiny, 1 thread)
// ---------------------------------------------------------------------------
__global__ void lg_scanB(int* __restrict__ blksums) {
  if (threadIdx.x == 0 && blockIdx.x == 0) {
    int run = 0;
    for (int b = 0; b < NBLK; ++b) {
      int t = blksums[b];
      blksums[b] = run;
      run += t;
    }
  }
}

// ---------------------------------------------------------------------------
// CSR build step 2c: add block offsets -> rowptr is final CSR offsets
// ---------------------------------------------------------------------------
__global__ void lg_scanC(int* __restrict__ rowptr, const int* __restrict__ blksums) {
  int i = blockIdx.x * blockDim.x + threadIdx.x;
  if (i >= NP1) return;
  rowptr[i] += blksums[i >> 10];
}

// ---------------------------------------------------------------------------
// CSR build step 3: scatter (val,col) into row-sorted order.
// adj triple staged into LDS with async global->LDS B128 DMA (CDNA5 path).
// ---------------------------------------------------------------------------
__global__ void lg_scatter(const float* __restrict__ vals, const int* __restrict__ rows,
                           const int* __restrict__ cols, int* __restrict__ rowcur,
                           float* __restrict__ svals, int* __restrict__ scols) {
  __shared__ float s_val[CHUNK];
  __shared__ int   s_row[CHUNK];
  __shared__ int   s_col[CHUNK];

  const int tid = threadIdx.x;  // 0..255
  const long long base = (long long)blockIdx.x * CHUNK;
  long long rem = (long long)NNZ - base;
  const int count = (rem < (long long)CHUNK) ? (int)rem : CHUNK;

  const int off = tid * 4;  // 16B granule inside chunk
#if ASYNC_LDS
  if (off < count) {  // NNZ % 4 == 0 so granules never straddle the end
    __builtin_amdgcn_global_load_async_to_lds_b128(
        (GV4*)(vals + base + off), (LV4*)&s_val[off], 0, 0);
    __builtin_amdgcn_global_load_async_to_lds_b128(
        (GV4*)(rows + base + off), (LV4*)&s_row[off], 0, 0);
    __builtin_amdgcn_global_load_async_to_lds_b128(
        (GV4*)(cols + base + off), (LV4*)&s_col[off], 0, 0);
  }
#if __has_builtin(__builtin_amdgcn_s_wait_asynccnt)
  __builtin_amdgcn_s_wait_asynccnt(0);
#else
  asm volatile("s_wait_asynccnt 0" ::: "memory");
#endif
#else
  if (off < count) {
    *(float4*)&s_val[off] = *(const float4*)(vals + base + off);
    *(int4*)&s_row[off]   = *(const int4*)(rows + base + off);
    *(int4*)&s_col[off]   = *(const int4*)(cols + base + off);
  }
#endif
  __syncthreads();

#pragma unroll
  for (int k = 0; k < 4; ++k) {
    int j = off + k;
    if (j < count) {
      int r = s_row[j];
      int p = atomicAdd(&rowcur[r], 1);
      svals[p] = s_val[j];
      scols[p] = s_col[j];
    }
  }
}

// ---------------------------------------------------------------------------
// Row-parallel CSR SpMM, atomic-free, with fused acc += result.
// One wave per row; lane owns 2 embedding elems; (val,col) read coalesced
// in 32-wide chunks and broadcast via wave32 shuffles.
// ---------------------------------------------------------------------------
__global__ void lg_spmm_csr(const int* __restrict__ rowptr,
                            const float* __restrict__ svals,
                            const int* __restrict__ scols,
                            const float* __restrict__ cur,
                            float* __restrict__ nxt,
                            float* __restrict__ acc) {
  int g = blockIdx.x * blockDim.x + threadIdx.x;
  int row  = g >> 5;
  int lane = g & 31;
  if (row >= N_NODES) return;
  const int e0 = lane * 2;

  int start = rowptr[row];
  int end   = rowptr[row + 1];

  float sx = 0.f, sy = 0.f;
  for (int p = start; p < end; p += 32) {
    int idx = p + lane;
    float v = 0.f;
    int   c = 0;
    if (idx < end) { v = svals[idx]; c = scols[idx]; }
    int m = end - p;
    if (m > 32) m = 32;
    for (int k = 0; k < m; ++k) {
      float vk = __shfl(v, k, 32);
      int   ck = __shfl(c, k, 32);
      const float2 x = *(const float2*)(cur + (long long)ck * EMB + e0);
      sx += vk * x.x;
      sy += vk * x.y;
    }
  }

  long long o = (long long)row * EMB + e0;
  float2 s; s.x = sx; s.y = sy;
  *(float2*)(nxt + o) = s;
  float2 a = *(const float2*)(acc + o);
  a.x += sx; a.y += sy;
  *(float2*)(acc + o) = a;
}

// ---------------------------------------------------------------------------
// Final gather + /4 scale.
// WMMA version: per wave, one 16x16 output tile computed as
//   D = (0.25 * I16) x X  via 4 chained V_WMMA_F32_16X16X4_F32.
// Exact (0.25 is a power of two; single nonzero product per K-sum) and robust
// to the (VGPR,half)->K slot convention since A and B share the same mapping.
// ---------------------------------------------------------------------------
#if HAVE_WMMA
__global__ void lg_gather(const float* __restrict__ acc, const int* __restrict__ users,
                          const int* __restrict__ pos, const int* __restrict__ neg,
                          float* __restrict__ out) {
  int g    = blockIdx.x * blockDim.x + threadIdx.x;
  int wave = g >> 5;          // wave id 0..12287 (launch has 4x the 3072 tiles; no guard)
  int lane = g & 31;
  int tileM = wave >> 2;      // 768 groups of 16 output rows
  int tileN = wave & 3;       // 4 groups of 16 embedding cols
  int rbase = tileM << 4;
  int cb    = tileN << 4;
  int which = rbase / BATCH;  // whole tile inside one segment (BATCH % 16 == 0)
  int bbase = rbase - which * BATCH;
  int n  = lane & 15;         // N column (B) / M row (A) held by this lane
  int hi = lane >> 4;

  v8f c = {};
#pragma unroll
  for (int b = 0; b < 4; ++b) {
    // logical K rows this lane's two VGPR slots hold (same mapping for A & B)
    int k0 = (hi ? 2 : 0);
    int k1 = (hi ? 3 : 1);
    // A block = 0.25 * I16[:, 4b..4b+3]  (lane holds M = n)
    v2f a;
    a.x = (n == 4 * b + k0) ? 0.25f : 0.0f;
    a.y = (n == 4 * b + k1) ? 0.25f : 0.0f;
    // B block = X rows 4b+k0 / 4b+k1, cols cb..cb+15 (gathered acc rows)
    int r0 = bbase + 4 * b + k0;
    int r1 = bbase + 4 * b + k1;
    int node0 = (which == 0) ? users[r0] : (which == 1) ? (pos[r0] + N_USER)
                                                        : (neg[r0] + N_USER);
    int node1 = (which == 0) ? users[r1] : (which == 1) ? (pos[r1] + N_USER)
                                                        : (neg[r1] + N_USER);
    v2f bb;
    bb.x = acc[(long long)node0 * EMB + cb + n];
    bb.y = acc[(long long)node1 * EMB + cb + n];
    c = __builtin_amdgcn_wmma_f32_16x16x4_f32(false, a, false, bb,
                                              (short)0, c, false, false);
  }
  // D layout: VGPR r, lanes 0-15 -> M=r, lanes 16-31 -> M=r+8; N = lane&15
#pragma unroll
  for (int r = 0; r < 8; ++r) {
    int orow = rbase + r + (hi ? 8 : 0);
    out[(long long)orow * EMB + cb + n] = c[r];
  }
}
#else
__global__ void lg_gather(const float* __restrict__ acc, const int* __restrict__ users,
                          const int* __restrict__ pos, const int* __restrict__ neg,
                          float* __restrict__ out) {
  int g = blockIdx.x * blockDim.x + threadIdx.x;
  int r = g >> 5;
  int lane = g & 31;
  if (r >= 3 * BATCH) return;
  int which = r / BATCH;
  int b = r - which * BATCH;
  int node = (which == 0) ? users[b]
           : (which == 1) ? (pos[b] + N_USER)
                          : (neg[b] + N_USER);
  const float2 v = *(const float2*)(acc + (long long)node * EMB + lane * 2);
  float2 o;
  o.x = v.x * 0.25f;
  o.y = v.y * 0.25f;
  *(float2*)(out + (long long)r * EMB + lane * 2) = o;
}
#endif

// ---------------------------------------------------------------------------
extern "C" void kernel_launch(void* const* d_in, const int* in_sizes, int n_in,
                              void* d_out, int out_size, void* d_ws, size_t ws_size,
                              hipStream_t stream) {
  const float* ue   = (const float*)d_in[0];
  const float* ie   = (const float*)d_in[1];
  const float* vals = (const float*)d_in[2];
  const int*   rows = (const int*)d_in[3];
  const int*   cols = (const int*)d_in[4];
  const int*   usr  = (const int*)d_in[5];
  const int*   pos  = (const int*)d_in[6];
  const int*   neg  = (const int*)d_in[7];
  float* out = (float*)d_out;

  // ---- workspace layout (all regions 256B-aligned by construction) ----
  const size_t embBytes = (size_t)N_NODES * EMB * sizeof(float);  // 38.4 MB
  const size_t nnzBytes = (size_t)NNZ * sizeof(float);            // 12 MB
  const size_t ptrBytes = (size_t)150016 * sizeof(int);           // rowptr region
  char* p = (char*)d_ws;
  float* acc    = (float*)p;             p += embBytes;
  float* curA   = (float*)p;             p += embBytes;
  float* curB   = (float*)p;             p += embBytes;
  float* svals  = (float*)p;             p += nnzBytes;
  int*   scols  = (int*)p;               p += nnzBytes;
  int*   rowptr = (int*)p;               p += ptrBytes;
  int*   rowcur = (int*)p;               p += ptrBytes;
  int*   blksums= (int*)p;               /* 256 ints */

  dim3 blk(256);
  const int total4 = N_NODES * EMB / 4;  // 2,400,000 float4 elements

  // 1) acc = cur = ego
  lg_init<<<(total4 + 255) / 256, blk, 0, stream>>>(ue, ie, acc, curA);

  // 2) CSR build: histogram -> scan -> scatter
  (void)hipMemsetAsync(rowptr, 0, ptrBytes, stream);
  lg_hist<<<(NNZ + 255) / 256, blk, 0, stream>>>(rows, rowptr);
  lg_scanA<<<NBLK, dim3(1024), 0, stream>>>(rowptr, blksums);
  lg_scanB<<<1, dim3(32), 0, stream>>>(blksums);
  lg_scanC<<<(NP1 + 255) / 256, blk, 0, stream>>>(rowptr, blksums);
  (void)hipMemcpyAsync(rowcur, rowptr, (size_t)N_NODES * sizeof(int),
                       hipMemcpyDeviceToDevice, stream);
  lg_scatter<<<(NNZ + CHUNK - 1) / CHUNK, blk, 0, stream>>>(vals, rows, cols,
                                                            rowcur, svals, scols);

  // 3) three propagation layers: atomic-free row-parallel SpMM + fused acc
  float* cur = curA;
  float* nxt = curB;
  for (int l = 0; l < N_LAYERS; ++l) {
    lg_spmm_csr<<<(N_NODES * 32) / 256, blk, 0, stream>>>(rowptr, svals, scols,
                                                          cur, nxt, acc);
    float* t = cur; cur = nxt; nxt = t;
  }

  // 4) batched gather with /4 scale (WMMA identity-scale tiles, exact)
  lg_gather<<<(3 * BATCH * 32) / 256, blk, 0, stream>>>(acc, usr, pos, neg, out);
}